// _CWNDefaultSecondConv_66511863546445
// MI455X (gfx1250) — compile-verified
//
#include <hip/hip_runtime.h>
#include <math.h>

// Problem constants (match reference)
#define N0V   100000
#define N1V   200000
#define NNZ_E 400000
#define DIM   256

// GEMM tiling
#define MT_PER_BLOCK 8   // one 16-row M-tile per wave, 8 waves per block
#define NP 64            // N panel per block (4 WMMA N-tiles per wave)
#define KSTR 260         // padded K stride of transposed LDS panel (floats)

typedef __attribute__((ext_vector_type(2))) float v2f;
typedef __attribute__((ext_vector_type(8))) float v8f;

// ---------------------------------------------------------------------------
// Zero the accumulation buffer (harness poisons d_out; atomics need zeros).
// ---------------------------------------------------------------------------
__global__ void zero_f4(float4* __restrict__ out, int n4) {
    int i = blockIdx.x * blockDim.x + threadIdx.x;
    if (i < n4) out[i] = make_float4(0.f, 0.f, 0.f, 0.f);
}

// ---------------------------------------------------------------------------
// msg = x_0 @ W  via V_WMMA_F32_16X16X4_F32 (fp32 matrix pipe).
//
// grid.x = ceil((N0/16)/8), grid.y = 256/64.  Block = 256 threads = 8 wave32.
// The block stages a [64 N x 256 K] *transposed* weight panel in LDS
// (stride 260 -> conflict-free b64 reads), then each wave owns one 16-row
// M-tile and produces a 16x64 stripe with 4 fp32 accumulator tiles.
//
// Per K-step (K+=4): 1 global b64 (A frag) + 4 ds_load_b64 (B frags, each
// landing directly in an even-aligned VGPR pair -> no repacking movs) + 4
// v_wmma_f32_16x16x4_f32.
//
// ISA layouts (CDNA5 ISA §7.12.2, wave32):
//   A 16x4 f32 : lanes 0-15 hold (M=lane, K, K+1); lanes 16-31 (M=lane-16, K+2, K+3)
//   B 4x16 f32 : lanes 0-15 hold (K, N=lane), (K+1, N=lane); lanes 16-31 K+2/K+3
//   C/D 16x16  : VGPR j: lanes 0-15 -> M=j, N=lane; lanes 16-31 -> M=8+j
// ---------------------------------------------------------------------------
__global__ __launch_bounds__(256) void gemm_wmma_f32(
    const float* __restrict__ x0,   // [N0][256]
    const float* __restrict__ w,    // [256][256]
    float* __restrict__ msg)        // [N0][256]
{
    __shared__ float wlds[NP * KSTR];           // 66,560 B transposed panel [n][k]

    const int tid  = threadIdx.x;
    const int lane = tid & 31;
    const int wave = tid >> 5;
    const int n0   = blockIdx.y * NP;

    // Stage weight panel transposed: wlds[n][k] = w[k][n0+n].
    // Coalesced float4 global reads along N, 4 scalar ds stores each.
    for (int i = tid; i < DIM * (NP / 4); i += 256) {
        int k  = i >> 4;              // 16 column-quads per K row
        int nq = (i & 15) << 2;
        float4 q = *(const float4*)&w[(size_t)k * DIM + n0 + nq];
        wlds[(nq + 0) * KSTR + k] = q.x;
        wlds[(nq + 1) * KSTR + k] = q.y;
        wlds[(nq + 2) * KSTR + k] = q.z;
        wlds[(nq + 3) * KSTR + k] = q.w;
    }
    __syncthreads();

    const int mtile = blockIdx.x * MT_PER_BLOCK + wave;
    if (mtile * 16 >= N0V) return;              // whole-wave guard (EXEC stays all-1s)
    const int  m0   = mtile * 16;
    const bool hi   = lane >= 16;
    const int  half = lane & 15;
    const int  koff = hi ? 2 : 0;

    const float* aptr  = x0 + (size_t)(m0 + half) * DIM + koff;
    const float* bbase = &wlds[half * KSTR + koff];   // this lane's B column

    v8f acc0 = {}, acc1 = {}, acc2 = {}, acc3 = {};

    #pragma unroll 4
    for (int k = 0; k < DIM; k += 4) {
        v2f a  = *(const v2f*)(aptr + k);               // A frag (global b64)
        v2f b0 = *(const v2f*)(bbase + k);              // B frags (ds b64, pre-paired)
        v2f b1 = *(const v2f*)(bbase + 16 * KSTR + k);
        v2f b2 = *(const v2f*)(bbase + 32 * KSTR + k);
        v2f b3 = *(const v2f*)(bbase + 48 * KSTR + k);

        acc0 = __builtin_amdgcn_wmma_f32_16x16x4_f32(false, a, false, b0, (short)0, acc0, false, false);
        acc1 = __builtin_amdgcn_wmma_f32_16x16x4_f32(false, a, false, b1, (short)0, acc1, false, false);
        acc2 = __builtin_amdgcn_wmma_f32_16x16x4_f32(false, a, false, b2, (short)0, acc2, false, false);
        acc3 = __builtin_amdgcn_wmma_f32_16x16x4_f32(false, a, false, b3, (short)0, acc3, false, false);
    }

    // Write back: VGPR j -> row m0 + j (lo half) / m0 + 8 + j (hi half).
    const int rbase = m0 + (hi ? 8 : 0);
    const int col   = n0 + half;
    #pragma unroll
    for (int j = 0; j < 8; ++j) {
        float* op = msg + (size_t)(rbase + j) * DIM + col;
        op[0]  = acc0[j];
        op[16] = acc1[j];
        op[32] = acc2[j];
        op[48] = acc3[j];
    }
}

// ---------------------------------------------------------------------------
// Edge scatter: out[rows[e]] += vals[e] * msg[cols[e]]  (fp32 L2 atomics).
// One wave32 per edge; each lane handles 8 floats as two coalesced float4s.
// ---------------------------------------------------------------------------
__global__ __launch_bounds__(256) void scatter_edges(
    const float* __restrict__ msg,
    const int*   __restrict__ rows,
    const int*   __restrict__ cols,
    const float* __restrict__ vals,
    float*       __restrict__ out)
{
    int edge = blockIdx.x * 8 + (threadIdx.x >> 5);
    if (edge >= NNZ_E) return;
    int lane = threadIdx.x & 31;

    int   r = rows[edge];
    int   c = cols[edge];
    float v = vals[edge];

    const float* src = msg + (size_t)c * DIM;
    float*       dst = out + (size_t)r * DIM;

    #pragma unroll
    for (int h = 0; h < 2; ++h) {
        int o = h * 128 + lane * 4;
        float4 m = *(const float4*)(src + o);
        unsafeAtomicAdd(dst + o + 0, m.x * v);
        unsafeAtomicAdd(dst + o + 1, m.y * v);
        unsafeAtomicAdd(dst + o + 2, m.z * v);
        unsafeAtomicAdd(dst + o + 3, m.w * v);
    }
}

// ---------------------------------------------------------------------------
// ELU in place over d_out.
// ---------------------------------------------------------------------------
__global__ void elu_f4(float4* __restrict__ out, int n4) {
    int i = blockIdx.x * blockDim.x + threadIdx.x;
    if (i < n4) {
        float4 t = out[i];
        t.x = t.x > 0.f ? t.x : expm1f(t.x);
        t.y = t.y > 0.f ? t.y : expm1f(t.y);
        t.z = t.z > 0.f ? t.z : expm1f(t.z);
        t.w = t.w > 0.f ? t.w : expm1f(t.w);
        out[i] = t;
    }
}

// ---------------------------------------------------------------------------
extern "C" void kernel_launch(void* const* d_in, const int* in_sizes, int n_in,
                              void* d_out, int out_size, void* d_ws, size_t ws_size,
                              hipStream_t stream) {
    const float* x0   = (const float*)d_in[0];
    // d_in[1] = x_1 : unused by the reference (update_func=None)
    const int*   rows = (const int*)  d_in[2];
    const int*   cols = (const int*)  d_in[3];
    const float* vals = (const float*)d_in[4];
    const float* wgt  = (const float*)d_in[5];

    float* out = (float*)d_out;
    float* msg = (float*)d_ws;               // N0*256 floats = 102.4 MB scratch

    const int out_elems = N1V * DIM;         // 51,200,000
    const int n4 = out_elems / 4;

    // 1) zero accumulator
    zero_f4<<<(n4 + 255) / 256, 256, 0, stream>>>((float4*)out, n4);

    // 2) msg = x0 @ W  (WMMA fp32)
    dim3 ggrid((N0V / 16 + MT_PER_BLOCK - 1) / MT_PER_BLOCK, DIM / NP);
    gemm_wmma_f32<<<ggrid, 256, 0, stream>>>(x0, wgt, msg);

    // 3) scatter-add edges
    scatter_edges<<<(NNZ_E + 7) / 8, 256, 0, stream>>>(msg, rows, cols, vals, out);

    // 4) ELU
    elu_f4<<<(n4 + 255) / 256, 256, 0, stream>>>((float4*)out, n4);
}